// SpecInverse1d_48850958024781
// MI455X (gfx1250) — compile-verified
//
#include <hip/hip_runtime.h>
#include <math.h>

// ---------------------------------------------------------------------------
// SpecInverse1d for MI455X (gfx1250, wave32)
//   Phase 1: serial order-4 recurrence (8188 steps) in one persistent WG,
//            LDS 5-slot row ring (80KB) + per-step wave32 shuffle reduction.
//   Phase 2: spectrum assembly ((out-b)/a, o prefix, phase, Hermitian mirror).
//   Phase 3: irfft(16384) as four-step FFT: two 128x128x128 complex matmuls
//            on V_WMMA_F32_16X16X4_F32 with twiddle between stages.
// Scratch layout in d_ws (320KB total):
//   [0      .. 64KB)  itout : 8192 x complex64 (recurrence dots, idx 4..8191 used)
//   [64KB   .. 192KB) Xf    : 16384 x complex64 (full Hermitian spectrum)
//   [192KB  .. 320KB) G     : 16384 x complex64 (stage-1 output, twiddled)
// ---------------------------------------------------------------------------

typedef float v2f __attribute__((ext_vector_type(2)));
typedef float v8f __attribute__((ext_vector_type(8)));

__device__ __forceinline__ float2 c_add(float2 a, float2 b) { return make_float2(a.x + b.x, a.y + b.y); }
__device__ __forceinline__ float2 c_sub(float2 a, float2 b) { return make_float2(a.x - b.x, a.y - b.y); }
__device__ __forceinline__ float2 c_mul(float2 a, float2 b) {
    return make_float2(a.x * b.x - a.y * b.y, a.x * b.y + a.y * b.x);
}

#ifndef M_PIf
#define M_PIf 3.14159265358979323846f
#endif

// ---------------------------------------------------------------------------
// Phase 1: P[k] = shift1(P[k-4]) - W0*P[k-4] - W3*P[k-1] - W2*P[k-2] - W1*P[k-3]
//          itout[k] = P[k] . v     (v = x.T.ravel(), width 2048)
// One workgroup, 1024 threads, 2 columns/thread, 1 barrier per step.
// ---------------------------------------------------------------------------
__global__ __launch_bounds__(1024)
void si_recurrence_kernel(const float2* __restrict__ x,   // (4,512) complex row-major
                          const float2* __restrict__ Wg,  // 4 complex
                          float2* __restrict__ itout)     // 8192 complex (writes 4..8191)
{
    extern __shared__ float2 smem[];
    float2* rows = smem;                 // 5 slots x 2048 complex = 80KB
    float2* part = smem + 5 * 2048;      // 2 x 32 complex (double-buffered partials)

    const int tid  = threadIdx.x;
    const int wave = tid >> 5;
    const int lane = tid & 31;
    const int i0 = tid;
    const int i1 = tid + 1024;

    const float2 W0 = Wg[0], W1 = Wg[1], W2 = Wg[2], W3 = Wg[3];
    // v[i] = x[i&3][i>>2]  (x.T.ravel of a (4,512) row-major array)
    const float2 v0 = x[(i0 & 3) * 512 + (i0 >> 2)];
    const float2 v1 = x[(i1 & 3) * 512 + (i1 >> 2)];

    // init P[0..3] = e_{512*r}
    for (int r = 0; r < 4; ++r) {
        rows[r * 2048 + i0] = make_float2((i0 == 512 * r) ? 1.f : 0.f, 0.f);
        rows[r * 2048 + i1] = make_float2((i1 == 512 * r) ? 1.f : 0.f, 0.f);
    }
    __syncthreads();

    for (int s = 4; s < 8192; ++s) {
        const int sl = s % 5;
        const float2* rm1 = rows + ((s + 4) % 5) * 2048;  // P[s-1]
        const float2* rm2 = rows + ((s + 3) % 5) * 2048;  // P[s-2]
        const float2* rm3 = rows + ((s + 2) % 5) * 2048;  // P[s-3]
        const float2* rm4 = rows + ((s + 1) % 5) * 2048;  // P[s-4]

        // column i0 (thread 0 gets the zero-fill of shift1)
        float2 sh0 = (i0 == 0) ? make_float2(0.f, 0.f) : rm4[i0 - 1];
        float2 n0  = c_sub(sh0, c_mul(W0, rm4[i0]));
        n0 = c_sub(n0, c_mul(W3, rm1[i0]));
        n0 = c_sub(n0, c_mul(W2, rm2[i0]));
        n0 = c_sub(n0, c_mul(W1, rm3[i0]));
        // column i1 (always > 0)
        float2 sh1 = rm4[i1 - 1];
        float2 n1  = c_sub(sh1, c_mul(W0, rm4[i1]));
        n1 = c_sub(n1, c_mul(W3, rm1[i1]));
        n1 = c_sub(n1, c_mul(W2, rm2[i1]));
        n1 = c_sub(n1, c_mul(W1, rm3[i1]));

        rows[sl * 2048 + i0] = n0;
        rows[sl * 2048 + i1] = n1;

        // partial dot with v, wave32 reduction
        float2 acc = c_add(c_mul(n0, v0), c_mul(n1, v1));
        for (int off = 16; off; off >>= 1) {
            acc.x += __shfl_xor(acc.x, off);
            acc.y += __shfl_xor(acc.y, off);
        }
        if (lane == 0) part[(s & 1) * 32 + wave] = acc;
        __syncthreads();

        if (tid < 32) {  // wave 0 finishes the cross-wave reduction
            float2 p = part[(s & 1) * 32 + tid];
            for (int off = 16; off; off >>= 1) {
                p.x += __shfl_xor(p.x, off);
                p.y += __shfl_xor(p.y, off);
            }
            if (tid == 0) itout[s] = p;
        }
    }
}

// ---------------------------------------------------------------------------
// Phase 2: X[0]=o ; X[j]=(it[j-1]-b)/a ; X *= exp(-2i*pi*h*j)
//          Xf[j]=X[j], Xf[16384-j]=conj(X[j]); imag of bins 0 and 8192 dropped.
// ---------------------------------------------------------------------------
__global__ void si_spectrum_kernel(const float2* __restrict__ o,
                                   const float* __restrict__ h,
                                   const float2* __restrict__ a,
                                   const float2* __restrict__ b,
                                   const float2* __restrict__ x,
                                   const float2* __restrict__ itout,
                                   float2* __restrict__ Xf)
{
    const int j = blockIdx.x * blockDim.x + threadIdx.x;
    if (j > 8192) return;

    float2 val;
    if (j == 0) {
        val = o[0];
    } else {
        float2 t;
        if (j <= 4) t = x[128 * (j - 1)];        // head: v[512*(j-1)] = x[0][128*(j-1)]
        else        t = itout[j - 1];
        const float2 num = c_sub(t, b[0]);
        const float2 ac  = a[0];
        const float inv  = 1.f / (ac.x * ac.x + ac.y * ac.y);
        val = make_float2((num.x * ac.x + num.y * ac.y) * inv,
                          (num.y * ac.x - num.x * ac.y) * inv);
    }
    float sn, cs;
    __sincosf(-2.f * M_PIf * h[0] * (float)j, &sn, &cs);
    val = c_mul(val, make_float2(cs, sn));
    if (j == 0 || j == 8192) val.y = 0.f;        // irfft ignores imag of DC/Nyquist
    Xf[j] = val;
    if (j >= 1 && j <= 8191) Xf[16384 - j] = make_float2(val.x, -val.y);
}

// ---------------------------------------------------------------------------
// Four-step inverse FFT, N = 128*128.  k = k1 + 128*k2, n = n2 + 128*n1.
// Stage 1: G[k1,n2]  = sum_k2 Xf[k1+128*k2] * e^{+2i*pi*k2*n2/128},
//          then twiddle by e^{+2i*pi*k1*n2/16384}.
// Stage 2: y[n2+128*n1] = Re( sum_k1 e^{+2i*pi*k1*n1/128} * G[k1,n2] ) / 16384
// One 16x16 output tile per wave; 64 tiles per stage.
// WMMA f32 16x16x4 fragment model (ISA 7.12.2):
//   A: lane -> M = lane&15, K = 2*(lane>=16) + {x:0, y:1}
//   B: lane -> N = lane&15, K = 2*(lane>=16) + {x:0, y:1}
//   C/D: vgpr r, lane -> M = r + 8*(lane>=16), N = lane&15
// ---------------------------------------------------------------------------
__device__ __forceinline__ v8f wmma_f32(v2f a, v2f b, v8f c) {
    return __builtin_amdgcn_wmma_f32_16x16x4_f32(false, a, false, b, (short)0, c, false, false);
}

__global__ __launch_bounds__(128)
void si_fft_stage1_kernel(const float2* __restrict__ Xf,  // 16384 complex
                          float2* __restrict__ G)         // 128x128 complex, [k1*128+n2]
{
    const int lane  = threadIdx.x & 31;
    const int tile  = blockIdx.x * (blockDim.x >> 5) + (threadIdx.x >> 5);  // 0..63
    const int tileM = tile >> 3;   // k1 / 16
    const int tileN = tile & 7;    // n2 / 16

    const int m   = lane & 15;
    const int kb  = (lane >> 4) * 2;       // 0 or 2
    const int k1A = tileM * 16 + m;        // A row (k1)
    const int n2B = tileN * 16 + m;        // B col (n2)

    v8f P1 = {}, P2 = {}, DI = {};
    for (int kc = 0; kc < 32; ++kc) {
        const int k2 = kc * 4 + kb;
        // A fragment: Xm[k1][k2] = Xf[k1 + 128*k2]  (coalesced over lanes)
        const float2 a0 = Xf[k1A + 128 * k2];
        const float2 a1 = Xf[k1A + 128 * (k2 + 1)];
        const v2f Ar = {a0.x, a1.x};
        const v2f Ai = {a0.y, a1.y};
        // B fragment: F[k2][n2] = e^{+2i*pi*k2*n2/128}
        float s0, c0, s1, c1;
        __sincosf((2.f * M_PIf / 128.f) * (float)((k2 * n2B) & 127), &s0, &c0);
        __sincosf((2.f * M_PIf / 128.f) * (float)(((k2 + 1) * n2B) & 127), &s1, &c1);
        const v2f Br = {c0, c1};
        const v2f Bi = {s0, s1};
        // complex MAC: Gr = sum ArBr - sum AiBi ; Gi = sum (ArBi + AiBr)
        P1 = wmma_f32(Ar, Br, P1);
        P2 = wmma_f32(Ai, Bi, P2);
        DI = wmma_f32(Ar, Bi, DI);
        DI = wmma_f32(Ai, Br, DI);
    }

    const int n2e = tileN * 16 + (lane & 15);
    const int mh  = (lane >= 16) ? 8 : 0;
    #pragma unroll
    for (int e = 0; e < 8; ++e) {
        const int k1e = tileM * 16 + e + mh;
        float gr = P1[e] - P2[e];
        float gi = DI[e];
        float ts, tc;  // twiddle e^{+2i*pi*k1*n2/16384}
        __sincosf((2.f * M_PIf / 16384.f) * (float)((k1e * n2e) & 16383), &ts, &tc);
        G[k1e * 128 + n2e] = make_float2(gr * tc - gi * ts, gr * ts + gi * tc);
    }
}

__global__ __launch_bounds__(128)
void si_fft_stage2_kernel(const float2* __restrict__ G,   // [k1*128+n2]
                          float* __restrict__ out)        // 16384 real
{
    const int lane  = threadIdx.x & 31;
    const int tile  = blockIdx.x * (blockDim.x >> 5) + (threadIdx.x >> 5);  // 0..63
    const int tileM = tile >> 3;   // n1 / 16
    const int tileN = tile & 7;    // n2 / 16

    const int m   = lane & 15;
    const int kb  = (lane >> 4) * 2;
    const int n1A = tileM * 16 + m;        // A row (n1)
    const int n2B = tileN * 16 + m;        // B col (n2)

    v8f P1 = {}, P2 = {};
    for (int kc = 0; kc < 32; ++kc) {
        const int k1 = kc * 4 + kb;
        // A fragment: e^{+2i*pi*n1*k1/128}
        float s0, c0, s1, c1;
        __sincosf((2.f * M_PIf / 128.f) * (float)((n1A * k1) & 127), &s0, &c0);
        __sincosf((2.f * M_PIf / 128.f) * (float)((n1A * (k1 + 1)) & 127), &s1, &c1);
        const v2f Ar = {c0, c1};
        const v2f Ai = {s0, s1};
        // B fragment: G[k1][n2]  (coalesced over lanes)
        const float2 b0 = G[k1 * 128 + n2B];
        const float2 b1 = G[(k1 + 1) * 128 + n2B];
        const v2f Br = {b0.x, b1.x};
        const v2f Bi = {b0.y, b1.y};
        // real part only: y = sum ArBr - sum AiBi
        P1 = wmma_f32(Ar, Br, P1);
        P2 = wmma_f32(Ai, Bi, P2);
    }

    const int n2e = tileN * 16 + (lane & 15);
    const int mh  = (lane >= 16) ? 8 : 0;
    #pragma unroll
    for (int e = 0; e < 8; ++e) {
        const int n1e = tileM * 16 + e + mh;
        out[n2e + 128 * n1e] = (P1[e] - P2[e]) * (1.f / 16384.f);
    }
}

// ---------------------------------------------------------------------------
extern "C" void kernel_launch(void* const* d_in, const int* in_sizes, int n_in,
                              void* d_out, int out_size, void* d_ws, size_t ws_size,
                              hipStream_t stream) {
    (void)in_sizes; (void)n_in; (void)out_size; (void)ws_size;
    const float2* o = (const float2*)d_in[0];
    const float*  h = (const float*) d_in[1];
    const float2* a = (const float2*)d_in[2];
    const float2* b = (const float2*)d_in[3];
    const float2* W = (const float2*)d_in[4];
    const float2* x = (const float2*)d_in[5];

    float2* itout = (float2*)d_ws;                            // 8192 c64
    float2* Xf    = (float2*)((char*)d_ws + 8192 * 8);        // 16384 c64
    float2* G     = Xf + 16384;                               // 16384 c64

    const size_t smem = (5 * 2048 + 64) * sizeof(float2);     // 82,432 B (<320KB/WGP)
    si_recurrence_kernel<<<1, 1024, smem, stream>>>(x, W, itout);
    si_spectrum_kernel<<<33, 256, 0, stream>>>(o, h, a, b, x, itout, Xf);
    si_fft_stage1_kernel<<<16, 128, 0, stream>>>(Xf, G);
    si_fft_stage2_kernel<<<16, 128, 0, stream>>>(G, (float*)d_out);
}